// AdapWingLoss_68985764708384
// MI455X (gfx1250) — compile-verified
//
#include <hip/hip_runtime.h>

// ---------------------------------------------------------------------------
// AdaptiveWingLoss, MI455X (gfx1250, wave32).
// Memory-bound streaming kernel: 134 MB in -> scalar out (~5.8us HBM floor).
// Uses V_WMMA_F32_16X16X4_F32 for the intra-wave reduction.
// ---------------------------------------------------------------------------

typedef float v2f __attribute__((ext_vector_type(2)));
typedef float v8f __attribute__((ext_vector_type(8)));

#define OMEGA 14.0f
#define ALPHA 2.1f
#define THETA 0.5f
#define LN2   0.69314718055994530942f

// Per-element adaptive wing loss. theta/eps == 0.5 exactly, so all powers of
// (theta/eps) collapse to exp2() — single v_exp_f32 / v_log_f32 ops.
__device__ __forceinline__ float awing(float x, float y) {
    float p      = ALPHA - y;                          // exponent, in (1.1, 2.1]
    float pow_te = __builtin_amdgcn_exp2f(-p);         // (0.5)^p
    float inv1pt = __builtin_amdgcn_rcpf(1.0f + pow_te);
    // A = omega * 1/(1+te^p) * p * te^(p-1) / eps ; te^(p-1) = 2*te^p
    float A = OMEGA * p * (2.0f * pow_te) * inv1pt;
    float log1p_te = __builtin_amdgcn_logf(1.0f + pow_te) * LN2;
    float C = THETA * A - OMEGA * log1p_te;
    float d = fabsf(y - x);
    // d^p = exp2(p * log2(d)); d==0 -> log2=-inf -> exp2(-inf)=0 (correct limit)
    float powd = __builtin_amdgcn_exp2f(p * __builtin_amdgcn_logf(d));
    float nl   = OMEGA * LN2 * __builtin_amdgcn_logf(1.0f + powd);
    float lin  = A * d - C;
    return d < THETA ? nl : lin;
}

// Wave32 sum-reduction via one WMMA:
//   A(16x4) : lane m (0..15) supplies A[m][0]=s, A[m][1]=0 in its 2 VGPRs;
//             lane m+16 supplies A[m][2]=s, A[m][3]=0.
//   B(4x16) : all-ones (layout-invariant).
//   D[m][n] = s_m + s_{m+16}  (row-sum, replicated over n).
// Summing the 8 D VGPRs gives lanes 0-15 -> sum(s_0..7,s_16..23),
// lanes 16-31 -> sum(s_8..15,s_24..31); one shfl_xor(16) completes the sum.
__device__ __forceinline__ float wave_reduce_sum(float s) {
    v2f a; a.x = s;    a.y = 0.0f;
    v2f b; b.x = 1.0f; b.y = 1.0f;
    v8f c = {};
    v8f d = __builtin_amdgcn_wmma_f32_16x16x4_f32(
        /*neg_a=*/false, a, /*neg_b=*/false, b,
        /*c_mod=*/(short)0, c, /*reuse_a=*/false, /*reuse_b=*/false);
    float t = (d[0] + d[1]) + (d[2] + d[3]) + (d[4] + d[5]) + (d[6] + d[7]);
    t += __shfl_xor(t, 16, 32);
    return t;   // full 32-lane sum, present in every lane
}

// Block reduction (256 threads = 8 waves on wave32). Deterministic tree order.
__device__ __forceinline__ float block_reduce_sum(float s) {
    __shared__ float wsum[8];
    const int lid = threadIdx.x & 31;
    const int wid = threadIdx.x >> 5;
    float w = wave_reduce_sum(s);
    if (lid == 0) wsum[wid] = w;
    __syncthreads();
    float v = 0.0f;
    if (wid == 0) {
        v = (lid < 8) ? wsum[lid] : 0.0f;
        v += __shfl_xor(v, 1, 32);
        v += __shfl_xor(v, 2, 32);
        v += __shfl_xor(v, 4, 32);
    }
    return v;   // valid in lane 0 of wave 0
}

__global__ void __launch_bounds__(256)
awing_partial_kernel(const float* __restrict__ x, const float* __restrict__ y,
                     float* __restrict__ part, int n) {
    const int tid    = blockIdx.x * blockDim.x + threadIdx.x;
    const int stride = gridDim.x * blockDim.x;
    const int n4     = n >> 2;

    const float4* __restrict__ x4 = (const float4*)x;
    const float4* __restrict__ y4 = (const float4*)y;

    float s = 0.0f;
    for (int i = tid; i < n4; i += stride) {            // global_load_b128 stream
        float4 xv = x4[i];
        float4 yv = y4[i];
        s += awing(xv.x, yv.x) + awing(xv.y, yv.y)
           + awing(xv.z, yv.z) + awing(xv.w, yv.w);
    }
    for (int i = (n4 << 2) + tid; i < n; i += stride)   // scalar tail (n%4)
        s += awing(x[i], y[i]);

    float bs = block_reduce_sum(s);
    if (threadIdx.x == 0) part[blockIdx.x] = bs;
}

__global__ void __launch_bounds__(256)
awing_final_kernel(const float* __restrict__ part, float* __restrict__ out, int nPart) {
    float s = 0.0f;
    for (int i = threadIdx.x; i < nPart; i += blockDim.x)
        s += part[i];
    float bs = block_reduce_sum(s);
    if (threadIdx.x == 0) out[0] = bs;
}

extern "C" void kernel_launch(void* const* d_in, const int* in_sizes, int n_in,
                              void* d_out, int out_size, void* d_ws, size_t ws_size,
                              hipStream_t stream) {
    const float* x = (const float*)d_in[0];   // input
    const float* y = (const float*)d_in[1];   // target
    float* out  = (float*)d_out;
    float* part = (float*)d_ws;

    const int n     = in_sizes[0];            // 32*2*512*512 = 16,777,216
    const int block = 256;                    // 8 waves/block (wave32)

    int maxBlocks = 4096;                     // 16 KB of d_ws partials
    size_t wsCap = ws_size / sizeof(float);
    if ((size_t)maxBlocks > wsCap && wsCap > 0) maxBlocks = (int)wsCap;
    if (maxBlocks < 1) maxBlocks = 1;

    int n4 = n >> 2;
    int blocks = (n4 + block - 1) / block;
    if (blocks > maxBlocks) blocks = maxBlocks;
    if (blocks < 1) blocks = 1;

    awing_partial_kernel<<<blocks, block, 0, stream>>>(x, y, part, n);
    awing_final_kernel<<<1, block, 0, stream>>>(part, out, blocks);
}